// LossFunction_13778255086185
// MI455X (gfx1250) — compile-verified
//
#include <hip/hip_runtime.h>

typedef float v2f __attribute__((ext_vector_type(2)));
typedef float v4f __attribute__((ext_vector_type(4)));
typedef float v8f __attribute__((ext_vector_type(8)));

#define TWO_PI_F     6.28318530717958647692f
#define INV_TWO_PI_F 0.15915494309189533577f
#define TT 2048   // T (time axis), power of two
#define FF 201    // F (freq axis)

// aw(x) = |x - rndne(x/2pi)*2pi|  (v_rndne_f32 == half-to-even, matches jnp.round)
__device__ __forceinline__ float anti_wrap(float x) {
  float r = rintf(x * INV_TWO_PI_F);
  return fabsf(fmaf(-r, TWO_PI_F, x));
}

__device__ __forceinline__ float wave_red(float v) {
#pragma unroll
  for (int off = 16; off > 0; off >>= 1) v += __shfl_xor(v, off, 32);
  return v;
}

// D = A x B + C with B = ones: every D[m,n] accumulates sum_k A[m,k].
// A vgpr0/vgpr1 carry one scalar each per lane -> 64 f32 adds per instruction
// on the matrix pipe, exact f32, deterministic.
__device__ __forceinline__ v8f wmma_acc2(float s0, float s1, v8f c) {
  v2f a; a[0] = s0; a[1] = s1;
  v2f b; b[0] = 1.0f; b[1] = 1.0f;
  return __builtin_amdgcn_wmma_f32_16x16x4_f32(false, a, false, b, (short)0, c,
                                               false, false);
}

// All 16 columns of D are identical (B==ones). Lane n (0-15) sums rows 0..7,
// lane n+16 sums rows 8..15; one xor-16 shuffle completes the 16-row total.
__device__ __forceinline__ float wmma_total(v8f d) {
  float s = d[0] + d[1] + d[2] + d[3] + d[4] + d[5] + d[6] + d[7];
  s += __shfl_xor(s, 16, 32);
  return s;
}

// ---------------------------------------------------------------------------
// Kernel A: loss_mag + 4 phase losses over [B,F,T] (float4 chunks, T%4==0 so
// a chunk never crosses a row; f-1 / t-1 neighbor reads hit L2).
// ---------------------------------------------------------------------------
__global__ __launch_bounds__(256) void mag_phase_kernel(
    const float* __restrict__ enhan_mag, const float* __restrict__ clean_mag,
    const float* __restrict__ clean_pha, float* __restrict__ wsA, int nChunks) {
  float s_mag = 0.f, s_ip = 0.f, s_gd = 0.f, s_iaf = 0.f, s_cspc = 0.f;
  const int tid = blockIdx.x * blockDim.x + threadIdx.x;
  const int stride = gridDim.x * blockDim.x;

  for (int c = tid; c < nChunks; c += stride) {
    const int i = c << 2;
    const int t0 = i & (TT - 1);
    const int f = (i >> 11) % FF;

    v4f em = *(const v4f*)(enhan_mag + i);
    v4f cm = __builtin_nontemporal_load((const v4f*)(clean_mag + i));
    v4f cp = *(const v4f*)(clean_pha + i);

    float d[4], aw[4];
#pragma unroll
    for (int j = 0; j < 4; j++) {
      float e = cm[j] - em[j];
      s_mag = fmaf(e, e, s_mag);
      d[j] = cp[j] - em[j];
    }
#pragma unroll
    for (int j = 0; j < 4; j++) {
      aw[j] = anti_wrap(d[j]);
      s_ip += aw[j];
      s_cspc += 1.0f - __cosf(aw[j]);
    }
    // group delay along F: gd = d[f-1] - d[f]; f==0 -> -d  (aw(-x)==aw(x))
    if (f != 0) {
      v4f emF = *(const v4f*)(enhan_mag + (i - TT));
      v4f cpF = *(const v4f*)(clean_pha + (i - TT));
#pragma unroll
      for (int j = 0; j < 4; j++) s_gd += anti_wrap((cpF[j] - emF[j]) - d[j]);
    } else {
#pragma unroll
      for (int j = 0; j < 4; j++) s_gd += aw[j];
    }
    // instantaneous freq along T: iaf = d[t-1] - d[t]; t==0 -> -d
    float dm1 = 0.0f;
    if (t0 != 0) dm1 = clean_pha[i - 1] - enhan_mag[i - 1];
    s_iaf += anti_wrap(dm1 - d[0]);
#pragma unroll
    for (int j = 1; j < 4; j++) s_iaf += anti_wrap(d[j - 1] - d[j]);
  }

  s_mag = wave_red(s_mag);
  s_ip = wave_red(s_ip);
  s_gd = wave_red(s_gd);
  s_iaf = wave_red(s_iaf);
  s_cspc = wave_red(s_cspc);

  __shared__ float red[8][5];
  const int wave = threadIdx.x >> 5, lane = threadIdx.x & 31;
  if (lane == 0) {
    red[wave][0] = s_mag; red[wave][1] = s_ip; red[wave][2] = s_gd;
    red[wave][3] = s_iaf; red[wave][4] = s_cspc;
  }
  __syncthreads();
  if (threadIdx.x == 0) {
    float o[5] = {0.f, 0.f, 0.f, 0.f, 0.f};
    for (int w = 0; w < 8; w++)
      for (int q = 0; q < 5; q++) o[q] += red[w][q];
    for (int q = 0; q < 5; q++) wsA[blockIdx.x * 5 + q] = o[q];
  }
}

// ---------------------------------------------------------------------------
// Kernel B: streaming sums for loss_com (sum sq-diff) and loss_time (sum abs
// diff), accumulated on the matrix pipe via v_wmma_f32_16x16x4_f32.
// NT loads: zero reuse, keep the 192MB L2 clean for kernel A's neighbors.
// ---------------------------------------------------------------------------
__global__ __launch_bounds__(256) void com_time_kernel(
    const float* __restrict__ enhan_com, const float* __restrict__ clean_com,
    const float* __restrict__ enhan_wav, const float* __restrict__ clean_wav,
    float* __restrict__ wsBC, int nCom4, int nWav4) {
  const int tid = blockIdx.x * blockDim.x + threadIdx.x;
  const int stride = gridDim.x * blockDim.x;
  v8f accC = {};
  v8f accT = {};

  const int itC = (nCom4 + 2 * stride - 1) / (2 * stride);  // uniform trip
  for (int k = 0; k < itC; k++) {
    const int c0 = k * 2 * stride + tid;
    const int c1 = c0 + stride;
    float s0 = 0.f, s1 = 0.f;
    if (c0 < nCom4) {
      v4f a = __builtin_nontemporal_load((const v4f*)enhan_com + c0);
      v4f b = __builtin_nontemporal_load((const v4f*)clean_com + c0);
      float e0 = b[0] - a[0], e1 = b[1] - a[1], e2 = b[2] - a[2], e3 = b[3] - a[3];
      s0 = fmaf(e0, e0, fmaf(e1, e1, fmaf(e2, e2, e3 * e3)));
    }
    if (c1 < nCom4) {
      v4f a = __builtin_nontemporal_load((const v4f*)enhan_com + c1);
      v4f b = __builtin_nontemporal_load((const v4f*)clean_com + c1);
      float e0 = b[0] - a[0], e1 = b[1] - a[1], e2 = b[2] - a[2], e3 = b[3] - a[3];
      s1 = fmaf(e0, e0, fmaf(e1, e1, fmaf(e2, e2, e3 * e3)));
    }
    accC = wmma_acc2(s0, s1, accC);  // EXEC all-ones here (uniform flow)
  }

  const int itT = (nWav4 + 2 * stride - 1) / (2 * stride);
  for (int k = 0; k < itT; k++) {
    const int c0 = k * 2 * stride + tid;
    const int c1 = c0 + stride;
    float s0 = 0.f, s1 = 0.f;
    if (c0 < nWav4) {
      v4f a = __builtin_nontemporal_load((const v4f*)enhan_wav + c0);
      v4f b = __builtin_nontemporal_load((const v4f*)clean_wav + c0);
      s0 = fabsf(b[0] - a[0]) + fabsf(b[1] - a[1]) + fabsf(b[2] - a[2]) +
           fabsf(b[3] - a[3]);
    }
    if (c1 < nWav4) {
      v4f a = __builtin_nontemporal_load((const v4f*)enhan_wav + c1);
      v4f b = __builtin_nontemporal_load((const v4f*)clean_wav + c1);
      s1 = fabsf(b[0] - a[0]) + fabsf(b[1] - a[1]) + fabsf(b[2] - a[2]) +
           fabsf(b[3] - a[3]);
    }
    accT = wmma_acc2(s0, s1, accT);
  }

  const float pC = wmma_total(accC);
  const float pT = wmma_total(accT);

  __shared__ float red[8][2];
  const int wave = threadIdx.x >> 5, lane = threadIdx.x & 31;
  if (lane == 0) { red[wave][0] = pC; red[wave][1] = pT; }
  __syncthreads();
  if (threadIdx.x == 0) {
    float a = 0.f, b = 0.f;
    for (int w = 0; w < 8; w++) { a += red[w][0]; b += red[w][1]; }
    wsBC[blockIdx.x * 2 + 0] = a;
    wsBC[blockIdx.x * 2 + 1] = b;
  }
}

// ---------------------------------------------------------------------------
// Finalize: one block; wave q reduces quantity q's block partials; wave 7
// does the 16-element metric loss. Thread 0 combines and writes d_out[0..5].
// ---------------------------------------------------------------------------
__global__ __launch_bounds__(256) void finalize_kernel(
    const float* __restrict__ wsA, const float* __restrict__ wsBC,
    const float* __restrict__ metric_g, const float* __restrict__ one_labels,
    float* __restrict__ out, int KA, int KBC, int N, int NW, int B) {
  __shared__ float sums[8];
  const int wave = threadIdx.x >> 5, lane = threadIdx.x & 31;
  float v = 0.f;
  if (wave < 5) {
    for (int b = lane; b < KA; b += 32) v += wsA[b * 5 + wave];
  } else if (wave < 7) {
    const int q = wave - 5;
    for (int b = lane; b < KBC; b += 32) v += wsBC[b * 2 + q];
  } else {
    if (lane < B) {
      float m = metric_g[lane] - one_labels[lane];
      v = m * m;
    }
  }
  v = wave_red(v);
  if (lane == 0) sums[wave] = v;
  __syncthreads();
  if (threadIdx.x == 0) {
    const float invN = 1.0f / (float)N;
    const float loss_mag = sums[0] * invN;
    const float loss_pha = (sums[1] + sums[2] + sums[3] + sums[4]) * invN;
    const float loss_com = sums[5] * invN;        // mean over 2N, *2 == /N
    const float loss_time = sums[6] / (float)NW;
    const float loss_metric = sums[7] / (float)B;
    const float nloss = loss_mag * 0.9f + loss_pha * 0.3f + loss_com * 0.1f +
                        loss_metric * 0.05f + loss_time * 0.2f;
    out[0] = nloss;
    out[1] = loss_mag;
    out[2] = loss_pha;
    out[3] = loss_com;
    out[4] = loss_metric;
    out[5] = loss_time;
  }
}

extern "C" void kernel_launch(void* const* d_in, const int* in_sizes, int n_in,
                              void* d_out, int out_size, void* d_ws,
                              size_t ws_size, hipStream_t stream) {
  (void)n_in; (void)out_size; (void)ws_size;
  const float* enhan_wav = (const float*)d_in[0];
  const float* enhan_mag = (const float*)d_in[1];
  /* d_in[2] = enhan_pha: unused by the reference (it passes enhan_mag). */
  const float* enhan_com = (const float*)d_in[3];
  const float* metric_g  = (const float*)d_in[4];
  const float* clean_wav = (const float*)d_in[5];
  const float* clean_mag = (const float*)d_in[6];
  const float* clean_pha = (const float*)d_in[7];
  const float* clean_com = (const float*)d_in[8];
  const float* one_labels = (const float*)d_in[9];

  const int N = in_sizes[1];          // B*F*T = 6,586,368
  const int nC4 = N >> 2;
  const int nCom4 = in_sizes[3] >> 2; // 2N/4
  const int nWav4 = in_sizes[0] >> 2; // B*L/4
  const int B = in_sizes[9];

  const int KA = 1024, KBC = 1024;
  float* wsA = (float*)d_ws;          // KA*5 floats
  float* wsBC = wsA + KA * 5;         // KBC*2 floats  (28 KB total)

  mag_phase_kernel<<<KA, 256, 0, stream>>>(enhan_mag, clean_mag, clean_pha,
                                           wsA, nC4);
  com_time_kernel<<<KBC, 256, 0, stream>>>(enhan_com, clean_com, enhan_wav,
                                           clean_wav, wsBC, nCom4, nWav4);
  finalize_kernel<<<1, 256, 0, stream>>>(wsA, wsBC, metric_g, one_labels,
                                         (float*)d_out, KA, KBC, N,
                                         in_sizes[0], B);
}